// ForwardWhLoss_25761213841996
// MI455X (gfx1250) — compile-verified
//
#include <hip/hip_runtime.h>

#define B_   128
#define H_   152
#define W_   272
#define HW_  (H_ * W_)
#define N_   500
#define NT   256   // 8 waves of 32 on CDNA5

typedef __attribute__((ext_vector_type(2))) float        v2f;
typedef __attribute__((ext_vector_type(8))) float        v8f;
typedef __attribute__((ext_vector_type(4))) unsigned int u32x4;
typedef __attribute__((ext_vector_type(8))) int          i32x8;
typedef __attribute__((ext_vector_type(4))) int          i32x4;

// One LDS struct so ids[] sits at LDS offset 0 (TDM destination).
struct SMem {
    int   ids[512];       // frame-1 ids, filled by tensor_load_to_lds
    float rxy[N_ * 2];    // scattered frame-2 xy (reordered_ind2)
    float rwh[N_ * 4];    // scattered frame-2 wh (reordered_wh2)
    float part[(NT / 32) * 3];
};

// Wave32 sum via V_WMMA_F32_16X16X4_F32: with B = all-ones,
// D[m,n] = rowsum(A[m,:]); column 0 (lanes 0 and 16, 8 VGPRs each)
// therefore holds all 16 row sums -> total = sum over all lanes' a[0].
__device__ __forceinline__ float wave_sum_wmma(float v) {
    v2f a; a[0] = v;    a[1] = 0.0f;
    v2f b; b[0] = 1.0f; b[1] = 1.0f;
    v8f c = {};
    c = __builtin_amdgcn_wmma_f32_16x16x4_f32(
            /*neg_a=*/false, a, /*neg_b=*/false, b,
            /*c_mod=*/(short)0, c, /*reuse_a=*/false, /*reuse_b=*/false);
    float s = c[0] + c[1] + c[2] + c[3] + c[4] + c[5] + c[6] + c[7];
    return __shfl(s, 0, 32) + __shfl(s, 16, 32);
}

__global__ __launch_bounds__(NT) void track_loss_kernel(
    const float* __restrict__ flow,  const float* __restrict__ p_wh,
    const int*   __restrict__ mask,  const int*   __restrict__ index,
    const int*   __restrict__ ids,   const float* __restrict__ wh,
    const int*   __restrict__ index2,const int*   __restrict__ ids2,
    const float* __restrict__ wh2,   float* __restrict__ partials)
{
    __shared__ SMem sm;
    const int b   = blockIdx.x;
    const int tid = threadIdx.x;

    // ---- Phase 0: zero scatter buffers while TDM DMAs ids[b,:] into LDS ----
    if (tid < 32) {   // wave 0 issues the tensor DMA (EXEC ignored by TDM)
        unsigned long long ga =
            (unsigned long long)(const void*)(ids + (size_t)b * N_);
        u32x4 g0;
        g0[0] = 1u;                                    // count = 1 (valid D#)
        g0[1] = 0u;                                    // lds_addr = 0 (sm.ids)
        g0[2] = (unsigned)(ga & 0xffffffffull);        // global_addr[31:0]
        g0[3] = (unsigned)((ga >> 32) & 0x01ffffffull) // global_addr[56:32]
              | (2u << 30);                            // type = 2 ("image")
        i32x8 g1;
        g1[0] = (2 << 16);                             // data_size = 4 bytes
        g1[1] = (int)((unsigned)(N_ & 0xffff) << 16);  // tensor_dim0[15:0]
        g1[2] = (N_ >> 16) | (1 << 16);                // dim0 hi | tensor_dim1=1
        g1[3] = (int)((unsigned)N_ << 16);             // tile_dim0 = 500
        g1[4] = 1;                                     // tile_dim1 = 1
        g1[5] = N_;                                    // tensor_dim0_stride
        g1[6] = 0;
        g1[7] = 0;
        i32x4 z4 = {};
#if defined(__clang_major__) && (__clang_major__ >= 23)
        i32x8 z8 = {};
        __builtin_amdgcn_tensor_load_to_lds(g0, g1, z4, z4, z8, 0);
#else
        __builtin_amdgcn_tensor_load_to_lds(g0, g1, z4, z4, 0);
#endif
        __builtin_amdgcn_s_wait_tensorcnt(0);
    }
    for (int i = tid; i < N_ * 2; i += NT) sm.rxy[i] = 0.0f;
    for (int i = tid; i < N_ * 4; i += NT) sm.rwh[i] = 0.0f;
    __syncthreads();

    // ---- Phase 1: O(N^2) id match (LDS-resident) + scatter into LDS ----
    for (int n2 = tid; n2 < N_; n2 += NT) {
        const int id2 = ids2[(size_t)b * N_ + n2];
        int cnt = 0, pos = 0;
#pragma unroll 4
        for (int j = 0; j < N_; ++j) {
            if (sm.ids[j] == id2) { ++cnt; if (cnt == 1) pos = j; }
        }
        if (cnt == 1 && id2 != 0) {
            const int   i2 = index2[(size_t)b * N_ + n2];
            sm.rxy[pos * 2 + 0] = (float)(i2 % W_);
            sm.rxy[pos * 2 + 1] = (float)i2 / (float)W_;
            const float* w2 = wh2 + ((size_t)b * N_ + n2) * 4;
            sm.rwh[pos * 4 + 0] = w2[0];
            sm.rwh[pos * 4 + 1] = w2[1];
            sm.rwh[pos * 4 + 2] = w2[2];
            sm.rwh[pos * 4 + 3] = w2[3];
        }
    }
    __syncthreads();

    // ---- Phase 2: gather flow/p_wh at index, accumulate L1 terms ----
    float axy = 0.0f, awh = 0.0f, am = 0.0f;
    const float* flowb = flow + (size_t)b * 2 * HW_;
    const float* pwhb  = p_wh + (size_t)b * 4 * HW_;
    for (int n = tid; n < N_; n += NT) {
        const float m  = (float)mask[(size_t)b * N_ + n];
        const int   ix = index[(size_t)b * N_ + n];
        const float xy0 = flowb[ix]       * m;
        const float xy1 = flowb[HW_ + ix] * m;
        axy += fabsf(xy0 - sm.rxy[n * 2 + 0])
             + fabsf(xy1 - sm.rxy[n * 2 + 1]);
        const float* whn = wh + ((size_t)b * N_ + n) * 4;
#pragma unroll
        for (int c4 = 0; c4 < 4; ++c4) {
            const float pw = pwhb[c4 * HW_ + ix] * m;
            const float w  = whn[c4];
            const float r  = sm.rwh[n * 4 + c4];
            float d = r - w;
            d = (d != -w) ? d : 0.0f;   // unmatched slot (r==0) -> zero
            d *= m;
            awh += fabsf(pw - d);
        }
        am += m;
    }

    // ---- Phase 3: wave reduce via WMMA, then fixed-order combine ----
    const float sxy = wave_sum_wmma(axy);
    const float swh = wave_sum_wmma(awh);
    const float smk = wave_sum_wmma(am);
    const int wid = tid >> 5;
    if ((tid & 31) == 0) {
        sm.part[wid * 3 + 0] = sxy;
        sm.part[wid * 3 + 1] = swh;
        sm.part[wid * 3 + 2] = smk;
    }
    __syncthreads();
    if (tid == 0) {
        float t0 = 0.0f, t1 = 0.0f, t2 = 0.0f;
        for (int wv = 0; wv < NT / 32; ++wv) {
            t0 += sm.part[wv * 3 + 0];
            t1 += sm.part[wv * 3 + 1];
            t2 += sm.part[wv * 3 + 2];
        }
        partials[b * 3 + 0] = t0;
        partials[b * 3 + 1] = t1;
        partials[b * 3 + 2] = t2;
    }
}

__global__ void finalize_kernel(const float* __restrict__ partials,
                                float* __restrict__ out)
{
    if (blockIdx.x == 0 && threadIdx.x == 0) {
        float sxy = 0.0f, swh = 0.0f, smk = 0.0f;
        for (int b = 0; b < B_; ++b) {       // fixed order -> deterministic
            sxy += partials[b * 3 + 0];
            swh += partials[b * 3 + 1];
            smk += partials[b * 3 + 2];
        }
        out[0] = sxy / (2.0f * smk + 1e-4f);   // mask2.sum() = 2*sum(mask)
        out[1] = swh / (4.0f * smk + 1e-4f);   // mask4.sum() = 4*sum(mask)
    }
}

extern "C" void kernel_launch(void* const* d_in, const int* in_sizes, int n_in,
                              void* d_out, int out_size, void* d_ws, size_t ws_size,
                              hipStream_t stream)
{
    const float* flow   = (const float*)d_in[0];
    const float* p_wh   = (const float*)d_in[1];
    const int*   mask   = (const int*)  d_in[2];
    const int*   index  = (const int*)  d_in[3];
    const int*   ids    = (const int*)  d_in[4];
    const float* wh     = (const float*)d_in[5];
    const int*   index2 = (const int*)  d_in[6];
    const int*   ids2   = (const int*)  d_in[7];
    const float* wh2    = (const float*)d_in[8];
    float* partials = (float*)d_ws;            // B_*3 floats = 1536 B

    track_loss_kernel<<<B_, NT, 0, stream>>>(flow, p_wh, mask, index, ids, wh,
                                             index2, ids2, wh2, partials);
    finalize_kernel<<<1, 32, 0, stream>>>(partials, (float*)d_out);
}